// BinaryClassifier_46909632807625
// MI455X (gfx1250) — compile-verified
//
#include <hip/hip_runtime.h>
#include <math.h>

typedef __attribute__((ext_vector_type(2))) float v2f;
typedef __attribute__((ext_vector_type(8))) float v8f;

#define EMB_DIM 100
#define SEQ_L   200
#define CHUNK   50
#define KSTEPS  25                    // 100 / 4
#define NTILES  ((SEQ_L + 15) / 16)   // 13
#define EPS_CS  1e-8f

// One wave32 per batch row b.
// WMMA computes, for 16 sequence positions at once, the 4 dot products
//   [dot(ch,u_norm0), dot(ch,u_norm1), dot(ch,w0), dot(ch,w1)]
// as a 16x100 @ 100x4 f32 GEMM via V_WMMA_F32_16X16X4_F32 (25 K-steps).
// B fragments are tile-invariant -> hoisted into registers once.
__global__ __launch_bounds__(32)
void attn_pool_wmma_kernel(const int* __restrict__ word_idxs,
                           const float* __restrict__ emb,
                           const float* __restrict__ weights,
                           const float* __restrict__ attend_u,
                           float* __restrict__ out, int B)
{
    __shared__ float V[EMB_DIM * 4];   // [elem][col], cols: u_n0|u_n1|w_lo|w_hi
    __shared__ float Dm[16 * 4];       // D-matrix cols 0..3 per row

    const int lane = threadIdx.x;      // 0..31 (wave32)
    const int b    = blockIdx.x;
    if (b >= B) return;
    const int n    = lane & 15;        // N / row-within-tile index
    const int koff = (lane >> 4) * 2;  // A/B fragment K sub-offset: 0 or 2

    // ---- query-vector norms (wave reduction) ----
    float su0 = 0.f, su1 = 0.f;
    for (int e = lane; e < CHUNK; e += 32) {
        float x0 = attend_u[e];
        float x1 = attend_u[CHUNK + e];
        su0 += x0 * x0;
        su1 += x1 * x1;
    }
    #pragma unroll
    for (int s = 16; s > 0; s >>= 1) {
        su0 += __shfl_xor(su0, s, 32);
        su1 += __shfl_xor(su1, s, 32);
    }
    const float inv_u0 = 1.f / fmaxf(sqrtf(su0), EPS_CS);
    const float inv_u1 = 1.f / fmaxf(sqrtf(su1), EPS_CS);

    // ---- build the 100x4 B-operand table in LDS ----
    for (int e = lane; e < EMB_DIM; e += 32) {
        float w  = weights[e];
        bool  lo = (e < CHUNK);
        V[e * 4 + 0] = lo ? attend_u[e] * inv_u0 : 0.f;  // u_norm chunk0 (padded)
        V[e * 4 + 1] = lo ? 0.f : attend_u[e] * inv_u1;  // u_norm chunk1 (attend_u flat)
        V[e * 4 + 2] = lo ? w : 0.f;                     // w chunk0
        V[e * 4 + 3] = lo ? 0.f : w;                     // w chunk1
    }
    __syncthreads();

    // ---- hoist all 25 B fragments into registers (tile-invariant) ----
    v2f bfrag[KSTEPS];
    #pragma unroll
    for (int k = 0; k < KSTEPS; ++k) {
        const int e0 = 4 * k + koff;
        if (n < 4) {
            bfrag[k].x = V[e0 * 4 + n];
            bfrag[k].y = V[(e0 + 1) * 4 + n];
        } else {
            bfrag[k].x = 0.f;
            bfrag[k].y = 0.f;
        }
    }

    const int* widx = word_idxs + (size_t)b * SEQ_L;

    // ---- prefetch all clamped gather indices (lane handles row n of each tile) ----
    int idxs[NTILES];
    #pragma unroll
    for (int t = 0; t < NTILES; ++t) {
        int l = t * 16 + n;
        idxs[t] = widx[(l < SEQ_L) ? l : (SEQ_L - 1)];
    }

    float num0 = 0.f, num1 = 0.f, den0 = 0.f, den1 = 0.f;

    for (int t = 0; t < NTILES; ++t) {
        const int l0 = t * 16;
        const float* rp = emb + (size_t)idxs[t] * EMB_DIM;

        v8f  c = {0.f, 0.f, 0.f, 0.f, 0.f, 0.f, 0.f, 0.f};
        float sq0 = 0.f, sq1 = 0.f;

        #pragma unroll
        for (int k = 0; k < KSTEPS; ++k) {
            const int e0 = 4 * k + koff;                 // even -> 8B aligned b64 load
            v2f a = *(const v2f*)(rp + e0);              // A frag: K={0,1}/{2,3}

            c = __builtin_amdgcn_wmma_f32_16x16x4_f32(
                    false, a, false, bfrag[k], (short)0, c, false, false);

            // per-chunk sum of squares rides along in VALU.
            // chunk boundary (elem 50) splits exactly at k==12 between lane halves.
            const float ss = a.x * a.x + a.y * a.y;
            if (k < 12)       sq0 += ss;
            else if (k > 12)  sq1 += ss;
            else              { if (lane < 16) sq0 += ss; else sq1 += ss; }
        }
        // combine the two lane-halves of each row's sum of squares
        sq0 += __shfl_xor(sq0, 16, 32);
        sq1 += __shfl_xor(sq1, 16, 32);

        // spill D cols 0..3 through LDS to get per-row scalars
        if (n < 4) {
            const int rbase = (lane >> 4) * 8;           // rows 0..7 / 8..15
            #pragma unroll
            for (int j = 0; j < 8; ++j)
                Dm[(rbase + j) * 4 + n] = c[j];
        }
        __syncthreads();

        if (lane < 16 && (l0 + lane) < SEQ_L) {
            const float du0 = Dm[lane * 4 + 0];
            const float du1 = Dm[lane * 4 + 1];
            const float dw0 = Dm[lane * 4 + 2];
            const float dw1 = Dm[lane * 4 + 3];
            const float ie0 = 1.f / fmaxf(sqrtf(sq0), EPS_CS);
            const float ie1 = 1.f / fmaxf(sqrtf(sq1), EPS_CS);
            const float a0  = __expf(du0 * ie0);         // alpha = exp(cos)
            const float a1  = __expf(du1 * ie1);
            den0 += a0;  den1 += a1;
            num0 += a0 * dw0;  num1 += a1 * dw1;
        }
        __syncthreads();
    }

    // wave reduction over sequence positions
    #pragma unroll
    for (int s = 16; s > 0; s >>= 1) {
        num0 += __shfl_xor(num0, s, 32);
        num1 += __shfl_xor(num1, s, 32);
        den0 += __shfl_xor(den0, s, 32);
        den1 += __shfl_xor(den1, s, 32);
    }
    if (lane == 0)
        out[b] = num0 / den0 + num1 / den1;
}

extern "C" void kernel_launch(void* const* d_in, const int* in_sizes, int n_in,
                              void* d_out, int out_size, void* d_ws, size_t ws_size,
                              hipStream_t stream) {
    (void)n_in; (void)d_ws; (void)ws_size; (void)out_size;
    const int*   word_idxs = (const int*)  d_in[0];   // [B, 200] int32
    const float* emb       = (const float*)d_in[1];   // [100000, 100] f32
    const float* weights   = (const float*)d_in[2];   // [100, 1] f32
    const float* attend_u  = (const float*)d_in[3];   // [2, 50] f32
    float*       out       = (float*)d_out;           // [B, 1] f32

    const int B = in_sizes[0] / SEQ_L;                // 4096
    attn_pool_wmma_kernel<<<B, 32, 0, stream>>>(word_idxs, emb, weights,
                                                attend_u, out, B);
}